// global_p_block_39015482917631
// MI455X (gfx1250) — compile-verified
//
#include <hip/hip_runtime.h>
#include <math.h>

typedef float v2f __attribute__((ext_vector_type(2)));
typedef float v8f __attribute__((ext_vector_type(8)));

#define P_ 31
#define N_ 128
#define C_ 128
#define S_ 60
#define B_ (N_ * S_)          // 7680
#define K_ 384                // C_*3
#define PC_ (P_ * C_)         // 3968
#define EPS_ 1e-5f
#define BN_RSQRT_ 0.99999500003749968f   // 1/sqrt(1+1e-5)
#define YSTRIDE 130           // padded LDS row stride (words); 130%64=2 -> conflict-free
#define YROWS 66              // rows 0 and 61..65 are zero pad (halo + s-tile overrun)

// ---------------------------------------------------------------------------
// K0: fold BatchNorm into grouped-conv weights, reorder K as dk*128 + c_in.
//   Wt[o][dk*128 + ci] = t_w[o][ci][dk] * alpha[o],  alpha = g*rsqrt(var+eps)
//   Beta[o] = b - alpha*mean
__global__ void wprep_kernel(const float* __restrict__ tw,
                             const float* __restrict__ g,
                             const float* __restrict__ bb,
                             const float* __restrict__ mean,
                             const float* __restrict__ var,
                             float* __restrict__ Wt,
                             float* __restrict__ Beta) {
  int o = blockIdx.x;           // p*128 + c_out, 0..3967
  int ci = threadIdx.x;         // 0..127
  float alpha = g[o] * rsqrtf(var[o] + EPS_);
  for (int dk = 0; dk < 3; ++dk)
    Wt[(size_t)o * K_ + dk * C_ + ci] = tw[((size_t)o * C_ + ci) * 3 + dk] * alpha;
  if (ci == 0) Beta[o] = bb[o] - alpha * mean[o];
}

// ---------------------------------------------------------------------------
// K1: channel-wise max + mean pooling.  PM/PA layout: [p][n*S + s]
__global__ void pool_kernel(const float* __restrict__ x,
                            float* __restrict__ PM, float* __restrict__ PA) {
  int pn = blockIdx.x;          // p*N + n
  int s = threadIdx.x;
  if (s >= S_) return;
  const float* xp = x + (size_t)pn * C_ * S_ + s;
  float mx = -__builtin_inff();
  float sm = 0.f;
  for (int c = 0; c < C_; ++c) {
    float v = xp[(size_t)c * S_];
    mx = fmaxf(mx, v);
    sm += v;
  }
  int idx = pn % N_;            // n
  int p = pn / N_;
  int o = p * B_ + idx * S_ + s;
  PM[o] = mx;
  PA[o] = sm * (1.0f / C_);
}

// ---------------------------------------------------------------------------
// K2: tiny ConvAttention + sigmoid gate. One wave (32 lanes) per b; lane = p.
__global__ void attn_gate_kernel(const float* __restrict__ PM,
                                 const float* __restrict__ PA,
                                 const float* __restrict__ dww,
                                 const float* __restrict__ dwb,
                                 const float* __restrict__ bng,
                                 const float* __restrict__ bnb,
                                 const float* __restrict__ pww,
                                 const float* __restrict__ pwb,
                                 const float* __restrict__ p2w,
                                 float* __restrict__ G) {
  __shared__ float kv[4][4][32];
  int lane = threadIdx.x & 31;
  int wv = threadIdx.x >> 5;
  int b = blockIdx.x * 4 + wv;
  int i = lane;
  bool act = (i < P_);
  float x0 = act ? PM[(size_t)i * B_ + b] : 0.f;   // channel 0 = max pool
  float x1 = act ? PA[(size_t)i * B_ + b] : 0.f;   // channel 1 = mean pool
  // neighbors with zero padding along p (width dim of the depthwise conv)
  float x0m = __shfl(x0, (lane + 31) & 31); if (i == 0) x0m = 0.f;
  float x0p = __shfl(x0, (lane + 1) & 31);  if (i >= P_ - 1) x0p = 0.f;
  float x1m = __shfl(x1, (lane + 31) & 31); if (i == 0) x1m = 0.f;
  float x1p = __shfl(x1, (lane + 1) & 31);  if (i >= P_ - 1) x1p = 0.f;

  float qkv[3][2];
#pragma unroll
  for (int t = 0; t < 3; ++t) {
    // depthwise 3x3 with H=1: only middle kernel row contributes
    const float* w0 = dww + ((t * 2 + 0) * 3 + 1) * 3;
    const float* w1 = dww + ((t * 2 + 1) * 3 + 1) * 3;
    float y0 = w0[0] * x0m + w0[1] * x0 + w0[2] * x0p + dwb[t * 2 + 0];
    float y1 = w1[0] * x1m + w1[1] * x1 + w1[2] * x1p + dwb[t * 2 + 1];
    // BN(eval, mean=0, var=1)
    y0 = bng[t * 2 + 0] * (y0 * BN_RSQRT_) + bnb[t * 2 + 0];
    y1 = bng[t * 2 + 1] * (y1 * BN_RSQRT_) + bnb[t * 2 + 1];
    // pointwise 2x2
    qkv[t][0] = pww[(t * 2 + 0) * 2 + 0] * y0 + pww[(t * 2 + 0) * 2 + 1] * y1 + pwb[t * 2 + 0];
    qkv[t][1] = pww[(t * 2 + 1) * 2 + 0] * y0 + pww[(t * 2 + 1) * 2 + 1] * y1 + pwb[t * 2 + 1];
  }
  kv[wv][0][lane] = qkv[1][0];   // k, d=0
  kv[wv][1][lane] = qkv[1][1];   // k, d=1
  kv[wv][2][lane] = qkv[2][0];   // v, d=0
  kv[wv][3][lane] = qkv[2][1];   // v, d=1
  __syncthreads();

  const float scale = 0.70710678118654752f;   // 2^-0.5
  float q0 = qkv[0][0] * scale, q1 = qkv[0][1] * scale;
  float mx = -__builtin_inff();
  for (int j = 0; j < P_; ++j)
    mx = fmaxf(mx, q0 * kv[wv][0][j] + q1 * kv[wv][1][j]);
  float sum = 0.f, o0 = 0.f, o1 = 0.f;
  for (int j = 0; j < P_; ++j) {
    float e = __expf(q0 * kv[wv][0][j] + q1 * kv[wv][1][j] - mx);
    sum += e;
    o0 += e * kv[wv][2][j];
    o1 += e * kv[wv][3][j];
  }
  float inv = 1.0f / sum;
  float tt = p2w[0] * (o0 * inv) + p2w[1] * (o1 * inv);
  float gate = 1.0f / (1.0f + __expf(-tt));
  if (act) G[(size_t)i * B_ + b] = gate;
}

// ---------------------------------------------------------------------------
// K3: grouped conv as 31 GEMMs (M=128, K=384, N=7680) using V_WMMA_F32_16X16X4_F32.
// One workgroup per (p,n); 8 waves -> 8 c_out tiles of 16; each wave holds
// 4 s-tile accumulators (s 0..63, store only s<60). Activations staged gated
// + transposed into LDS with +-1 halo so B fragments are contiguous b64 loads.
__global__ __launch_bounds__(256) void conv_gemm_kernel(
    const float* __restrict__ x, const float* __restrict__ G,
    const float* __restrict__ Wt, const float* __restrict__ Beta,
    float* __restrict__ out) {
  __shared__ float Yt[YROWS * YSTRIDE];
  int pn = blockIdx.x;                 // p*N + n (groups same-p blocks for L2 weight reuse)
  int p = pn / N_;
  int n = pn % N_;
  int tid = threadIdx.x;

  // zero rows 0 and 61..65 (conv halo + s-tile overrun)
  for (int idx = tid; idx < 6 * YSTRIDE; idx += 256) {
    int r = idx / YSTRIDE, c = idx % YSTRIDE;
    int row = (r == 0) ? 0 : (60 + r);
    Yt[row * YSTRIDE + c] = 0.f;
  }
  // stage gated activations: Yt[s+1][c] = (1 + gate[n,s,p]) * x[p,n,c,s]
  const float* xp = x + (size_t)pn * C_ * S_;
  const float* gp = G + (size_t)p * B_ + n * S_;
  for (int idx = tid; idx < C_ * S_; idx += 256) {   // 7680 = 30*256, no divergence
    int c = idx / S_, s = idx % S_;
    Yt[(s + 1) * YSTRIDE + c] = xp[idx] * (1.0f + gp[s]);
  }
  __syncthreads();

  int wv = tid >> 5, lane = tid & 31;
  int m = lane & 15;                   // A row (c_out) / B column (s)
  int hi = lane >> 4;
  int k0 = hi * 2;                     // K pair offset within K-group of 4
  int cob = wv * 16;                   // c_out tile base
  const float* wrow = Wt + (size_t)(p * C_ + cob + m) * K_;

  v8f acc0 = {0,0,0,0,0,0,0,0};
  v8f acc1 = {0,0,0,0,0,0,0,0};
  v8f acc2 = {0,0,0,0,0,0,0,0};
  v8f acc3 = {0,0,0,0,0,0,0,0};

  for (int kk = 0; kk < 96; ++kk) {
    int t0 = 4 * kk + k0;              // K index (dk uniform within each 4-group)
    int dk = t0 >> 7;                  // 0..2
    int c0 = t0 & 127;                 // c_in (even -> 8B aligned)
    v2f a = *reinterpret_cast<const v2f*>(wrow + t0);
    v2f b0 = *reinterpret_cast<const v2f*>(&Yt[( 0 + m + dk) * YSTRIDE + c0]);
    v2f b1 = *reinterpret_cast<const v2f*>(&Yt[(16 + m + dk) * YSTRIDE + c0]);
    v2f b2 = *reinterpret_cast<const v2f*>(&Yt[(32 + m + dk) * YSTRIDE + c0]);
    v2f b3 = *reinterpret_cast<const v2f*>(&Yt[(48 + m + dk) * YSTRIDE + c0]);
    acc0 = __builtin_amdgcn_wmma_f32_16x16x4_f32(false, a, false, b0, (short)0, acc0, false, false);
    acc1 = __builtin_amdgcn_wmma_f32_16x16x4_f32(false, a, false, b1, (short)0, acc1, false, false);
    acc2 = __builtin_amdgcn_wmma_f32_16x16x4_f32(false, a, false, b2, (short)0, acc2, false, false);
    acc3 = __builtin_amdgcn_wmma_f32_16x16x4_f32(false, a, false, b3, (short)0, acc3, false, false);
  }

  // epilogue: D VGPR r holds (M = r + 8*hi, N = m). Add folded BN bias, store.
  union { v8f v; float f[8]; } u0, u1, u2, u3;
  u0.v = acc0; u1.v = acc1; u2.v = acc2; u3.v = acc3;
  float* ob = out + (size_t)pn * C_ * S_;
#pragma unroll
  for (int r = 0; r < 8; ++r) {
    int cout = cob + hi * 8 + r;
    float beta = Beta[p * C_ + cout];
    float* orow = ob + (size_t)cout * S_;
    orow[ 0 + m] = u0.f[r] + beta;
    orow[16 + m] = u1.f[r] + beta;
    orow[32 + m] = u2.f[r] + beta;
    if (48 + m < S_) orow[48 + m] = u3.f[r] + beta;
  }
}

// ---------------------------------------------------------------------------
extern "C" void kernel_launch(void* const* d_in, const int* in_sizes, int n_in,
                              void* d_out, int out_size, void* d_ws, size_t ws_size,
                              hipStream_t stream) {
  const float* x   = (const float*)d_in[0];
  const float* dww = (const float*)d_in[1];
  const float* dwb = (const float*)d_in[2];
  const float* bng = (const float*)d_in[3];
  const float* bnb = (const float*)d_in[4];
  const float* pww = (const float*)d_in[5];
  const float* pwb = (const float*)d_in[6];
  const float* p2w = (const float*)d_in[7];
  const float* tw  = (const float*)d_in[8];
  const float* tg  = (const float*)d_in[9];
  const float* tb  = (const float*)d_in[10];
  const float* tm  = (const float*)d_in[11];
  const float* tv  = (const float*)d_in[12];
  float* out = (float*)d_out;

  char* ws = (char*)d_ws;
  float* Wt   = (float*)(ws);                 // 3968*384*4 = 6,094,848 B
  float* Beta = (float*)(ws + 6094848);       // 15,872 B
  float* PM   = (float*)(ws + 6110720);       // 952,320 B
  float* PA   = (float*)(ws + 7063040);       // 952,320 B
  float* G    = (float*)(ws + 8015360);       // 952,320 B  (total ~8.97 MB)

  wprep_kernel<<<PC_, C_, 0, stream>>>(tw, tg, tb, tm, tv, Wt, Beta);
  pool_kernel<<<P_ * N_, 64, 0, stream>>>(x, PM, PA);
  attn_gate_kernel<<<B_ / 4, 128, 0, stream>>>(PM, PA, dww, dwb, bng, bnb,
                                               pww, pwb, p2w, G);
  conv_gemm_kernel<<<P_ * N_, 256, 0, stream>>>(x, G, Wt, Beta, out);
}